// GRecLayer_1683627180108
// MI455X (gfx1250) — compile-verified
//
#include <hip/hip_runtime.h>
#include <stdint.h>

typedef __attribute__((ext_vector_type(2))) float v2f;
typedef __attribute__((ext_vector_type(8))) float v8f;
typedef __attribute__((ext_vector_type(4))) int   v4i;

#define AS1 __attribute__((address_space(1)))
#define AS3 __attribute__((address_space(3)))

#define N_NODES 50000
#define N_EDGES 800000
#define D 256
#define LEAKY 0.2f
#define KCHUNK 16               // K rows of W staged per LDS buffer
#define NCHUNK (D / KCHUNK)     // 16 chunks

#if __has_builtin(__builtin_amdgcn_global_load_async_to_lds_b128)
#define HAVE_ASYNC_LDS 1
#else
#define HAVE_ASYNC_LDS 0
#endif

__device__ __forceinline__ void wait_async_lds() {
#if HAVE_ASYNC_LDS
#if __has_builtin(__builtin_amdgcn_s_wait_asynccnt)
    __builtin_amdgcn_s_wait_asynccnt(0);
#else
    asm volatile("s_wait_asynccnt 0" ::: "memory");
#endif
#endif
}

// ---------------------------------------------------------------- kernel 1
// zero the ax accumulator (50000 x 256 f32) in workspace, float4 stores
__global__ void grec_zero_ax(float4* __restrict__ ax4) {
    int i = blockIdx.x * blockDim.x + threadIdx.x;   // 3.2M float4s exactly
    ax4[i] = make_float4(0.f, 0.f, 0.f, 0.f);
}

// ---------------------------------------------------------------- kernel 2
// SpMM, one wave per edge: edge scalars go through readfirstlane so they are
// wave-uniform (SGPR/SMEM path, loaded once per wave instead of per-lane).
// Each lane owns 8 features: 2x float4 gather + 8 global_atomic_add_f32.
// features (51.2MB) and ax (51.2MB) both stay resident in the 192MB L2.
__global__ __launch_bounds__(256) void grec_spmm(
    const float* __restrict__ feat,
    const long long* __restrict__ esrc,
    const long long* __restrict__ edst,
    const float* __restrict__ ew,
    float* __restrict__ ax) {
    const int lane = threadIdx.x & 31;
    const int e = __builtin_amdgcn_readfirstlane(blockIdx.x * 8 + (threadIdx.x >> 5));
    const int s = __builtin_amdgcn_readfirstlane((int)esrc[e]);
    const int d = __builtin_amdgcn_readfirstlane((int)edst[e]);
    const float w = ew[e];
    const int c = lane * 8;
    const float4 f0 = *reinterpret_cast<const float4*>(feat + (size_t)s * D + c);
    const float4 f1 = *reinterpret_cast<const float4*>(feat + (size_t)s * D + c + 4);
    float* p = ax + (size_t)d * D + c;
    atomicAdd(p + 0, f0.x * w);
    atomicAdd(p + 1, f0.y * w);
    atomicAdd(p + 2, f0.z * w);
    atomicAdd(p + 3, f0.w * w);
    atomicAdd(p + 4, f1.x * w);
    atomicAdd(p + 5, f1.y * w);
    atomicAdd(p + 6, f1.z * w);
    atomicAdd(p + 7, f1.w * w);
}

// ---------------------------------------------------------------- kernel 3
// fused GEMM: out = (ax + x) @ W1 + (ax * x) @ W2 + b, leaky-relu.
// fp32 WMMA (V_WMMA_F32_16X16X4_F32), one chained C accumulator.
// W1/W2 K-slices are double-buffered in LDS via GLOBAL_LOAD_ASYNC_TO_LDS_B128
// (ASYNCcnt + barrier sync), so all 8 waves of the block share one W stream.
__global__ __launch_bounds__(256) void grec_gemm(
    const float* __restrict__ feat, const float* __restrict__ ax,
    const float* __restrict__ W1,   const float* __restrict__ W2,
    const float* __restrict__ bias, float* __restrict__ out) {

    __shared__ float ldsW[2][2][KCHUNK * D];   // [stage][matrix][16*256] = 64KB

    const int tid   = threadIdx.x;
    const int lane  = tid & 31;
    const int wave  = tid >> 5;
    const int mbase = blockIdx.x * 32 + (wave >> 2) * 16;  // 16-row slab
    const int nbase = (wave & 3) * 64;                      // 64-col slab

    // A-fragment addressing (32-bit A 16x4): lanes 0-15 -> K pair {0,1},
    // lanes 16-31 -> K pair {2,3}; VGPR0 = even K, VGPR1 = odd K.
    const int m     = lane & 15;
    const int khalf = (lane >> 4) * 2;          // 0 or 2
    int row = mbase + m;
    if (row >= N_NODES) row = N_NODES - 1;      // clamp: EXEC must stay all-1s
    const float* axr = ax   + (size_t)row * D;
    const float* xr  = feat + (size_t)row * D;
    const int ncol = nbase + (lane & 15);       // B fragment: lane = N

    // cooperative prefetch of one 16-row K-slice of W1+W2 into stage st:
    // 2 matrices * 16*256 f32 = 2048 float4 slots, 8 per thread, b128 async.
    auto prefetch = [&](int kc, int st) {
        const size_t gbase = (size_t)kc * KCHUNK * D;
#pragma unroll
        for (int i = 0; i < 8; ++i) {
            const int idx = tid + i * 256;      // 0..2047
            const int mat = idx >> 10;          // constant after unroll
            const int rem = idx & 1023;         // float4 slot in 16x256 slab
            const float* gsrc = (mat ? W2 : W1) + gbase + rem * 4;
            float* ldst = &ldsW[st][mat][rem * 4];
#if HAVE_ASYNC_LDS
            __builtin_amdgcn_global_load_async_to_lds_b128(
                (AS1 v4i*)gsrc, (AS3 v4i*)ldst, 0, 0);
#else
            *reinterpret_cast<float4*>(ldst) =
                *reinterpret_cast<const float4*>(gsrc);
#endif
        }
    };

    v8f acc[4] = {};

    prefetch(0, 0);
    for (int kc = 0; kc < NCHUNK; ++kc) {
        wait_async_lds();          // this wave's async slice landed in LDS
        __syncthreads();           // everyone's slice landed; prev reads done
        if (kc + 1 < NCHUNK) prefetch(kc + 1, (kc + 1) & 1);  // wave-uniform

        const int st = kc & 1;
#pragma unroll
        for (int k4 = 0; k4 < KCHUNK / 4; ++k4) {
            const int klocal = k4 * 4 + khalf;       // this lane's K pair base
            const int kk = kc * KCHUNK + klocal;
            const float2 a = *reinterpret_cast<const float2*>(axr + kk);
            const float2 x = *reinterpret_cast<const float2*>(xr + kk);
            v2f Aagg = { a.x + x.x, a.y + x.y };     // (A + I) @ X path
            v2f Anbr = { a.x * x.x, a.y * x.y };     // (A @ X) * X path
            const float* l1 = &ldsW[st][0][klocal * D + ncol];
            const float* l2 = &ldsW[st][1][klocal * D + ncol];
#pragma unroll
            for (int nt = 0; nt < 4; ++nt) {
                v2f B1 = { l1[nt * 16], l1[D + nt * 16] };
                v2f B2 = { l2[nt * 16], l2[D + nt * 16] };
                acc[nt] = __builtin_amdgcn_wmma_f32_16x16x4_f32(
                    false, Aagg, false, B1, (short)0, acc[nt], false, false);
                acc[nt] = __builtin_amdgcn_wmma_f32_16x16x4_f32(
                    false, Anbr, false, B2, (short)0, acc[nt], false, false);
            }
        }
    }

    // C/D layout: VGPR r -> M = r (lanes 0-15) or r+8 (lanes 16-31), N = lane%16
    const int mtop = (lane >> 4) * 8;
#pragma unroll
    for (int nt = 0; nt < 4; ++nt) {
        const int ocol = nbase + nt * 16 + (lane & 15);
        const float bv = bias[ocol];
#pragma unroll
        for (int r = 0; r < 8; ++r) {
            const int orow = mbase + mtop + r;
            if (orow < N_NODES) {
                float v = acc[nt][r] + bv;
                out[(size_t)orow * D + ocol] = (v >= 0.f) ? v : LEAKY * v;
            }
        }
    }
}

// ---------------------------------------------------------------- launcher
extern "C" void kernel_launch(void* const* d_in, const int* in_sizes, int n_in,
                              void* d_out, int out_size, void* d_ws, size_t ws_size,
                              hipStream_t stream) {
    const float*     feat = (const float*)d_in[0];
    const long long* esrc = (const long long*)d_in[1];   // int64 in reference
    const long long* edst = (const long long*)d_in[2];
    const float*     ew   = (const float*)d_in[3];
    const float*     W1   = (const float*)d_in[4];
    const float*     W2   = (const float*)d_in[5];
    const float*     bias = (const float*)d_in[6];
    float* out = (float*)d_out;
    float* ax  = (float*)d_ws;   // 50000*256*4 = 51.2 MB scratch

    // 1) ax = 0
    grec_zero_ax<<<(N_NODES * D / 4) / 256, 256, 0, stream>>>((float4*)ax);
    // 2) ax += scatter_add(features[src] * w)  (one wave per edge)
    grec_spmm<<<N_EDGES / 8, 256, 0, stream>>>(feat, esrc, edst, ew, ax);
    // 3) out = leaky_relu((ax+x)@W1 + (ax*x)@W2 + b) via fp32 WMMA + async LDS
    grec_gemm<<<(N_NODES + 31) / 32, 256, 0, stream>>>(feat, ax, W1, W2, bias, out);
}